// GNNEval_7129645711376
// MI455X (gfx1250) — compile-verified
//
#include <hip/hip_runtime.h>

#define N_NODES  20000
#define N_EDGES  320000
#define N_GRAPHS 64
#define NODE_DIM 128
#define GLOB_DIM 16
#define HIDDEN   256

typedef __attribute__((ext_vector_type(2))) float v2f;
typedef __attribute__((ext_vector_type(8))) float v8f;

// ---------------- z = (1 + eps[layer]) * h ----------------
__global__ __launch_bounds__(256) void zinit_kernel(const float* __restrict__ h,
                                                    const float* __restrict__ eps, int layer,
                                                    float* __restrict__ z, int n) {
    int i = blockIdx.x * 256 + threadIdx.x;
    if (i < n) z[i] = (1.0f + eps[layer]) * h[i];
}

// ---------------- z[dst] += h[src]; one wave per edge ----------------
// Lane l moves floats [4l, 4l+4) of the feature row per iteration (128 floats / wave / iter).
__global__ __launch_bounds__(256) void scatter_kernel(const float* __restrict__ h,
                                                      const int* __restrict__ src,
                                                      const int* __restrict__ dst,
                                                      float* __restrict__ z,
                                                      int n_edges, int hid) {
    const int e    = (blockIdx.x * 256 + threadIdx.x) >> 5;   // edge = wave index
    const int lane = threadIdx.x & 31;
    if (e >= n_edges) return;
    const int s = src[e];
    const int d = dst[e];
    const float* __restrict__ hp = h + (size_t)s * hid;
    float* __restrict__ zp = z + (size_t)d * hid;
    for (int c = lane * 4; c < hid; c += 128) {
        const float4 v = *(const float4*)(hp + c);
        atomicAdd(zp + c + 0, v.x);
        atomicAdd(zp + c + 1, v.y);
        atomicAdd(zp + c + 2, v.z);
        atomicAdd(zp + c + 3, v.w);
    }
}

// ---- out = ReLU(A[MxK] @ W[KxN=256] + b) (+ residual) via V_WMMA_F32_16X16X4_F32 ----
// grid = M/16 blocks; block = 128 threads (4 waves). Wave w owns cols [64w, 64w+64) as four
// 16x16 accumulators; the A fragment is loaded once per K-step and reused across all 4 WMMAs.
template <int K, bool HAS_RESID>
__global__ __launch_bounds__(128) void gemm_relu_wmma(const float* __restrict__ A,
                                                      const float* __restrict__ W,
                                                      const float* __restrict__ bias,
                                                      const float* __restrict__ residual,
                                                      float* __restrict__ out) {
    constexpr int N = HIDDEN;
    const int lane  = threadIdx.x & 31;
    const int wave  = threadIdx.x >> 5;
    const int mbase = blockIdx.x << 4;
    const int nbase = wave << 6;          // 64-col strip per wave
    const int half  = lane >> 4;          // 0: lanes 0-15, 1: lanes 16-31
    const int l16   = lane & 15;
    const int koff  = half << 1;          // this half-wave covers K = kbase+{0,1} / kbase+{2,3}

    const float* __restrict__ Arow = A + (size_t)(mbase + l16) * K + koff;
    const float* __restrict__ Wp   = W + (size_t)koff * N + nbase + l16;

    v8f acc[4] = {};
#pragma unroll 2
    for (int k = 0; k < K; k += 4) {
        v2f a;
        a.x = Arow[k];
        a.y = Arow[k + 1];
#pragma unroll
        for (int j = 0; j < 4; ++j) {
            v2f b;
            b.x = Wp[(size_t)k * N + j * 16];
            b.y = Wp[(size_t)(k + 1) * N + j * 16];
            acc[j] = __builtin_amdgcn_wmma_f32_16x16x4_f32(false, a, false, b, (short)0,
                                                           acc[j], false, false);
        }
    }

#pragma unroll
    for (int j = 0; j < 4; ++j) {
        const int col = nbase + j * 16 + l16;
        const float bv = bias[col];
#pragma unroll
        for (int r = 0; r < 8; ++r) {
            const int orow = mbase + (half << 3) + r;   // D layout: VGPR r -> row r (+8 hi half)
            float v = fmaxf(acc[j][r] + bv, 0.0f);
            if (HAS_RESID) v += residual[(size_t)orow * N + col];
            out[(size_t)orow * N + col] = v;
        }
    }
}

// ---------------- small zero-fill ----------------
__global__ __launch_bounds__(256) void zero_kernel(float* __restrict__ p, int n) {
    int i = blockIdx.x * 256 + threadIdx.x;
    if (i < n) p[i] = 0.0f;
}

// ---------------- per-graph sums + counts ----------------
__global__ __launch_bounds__(256) void pool_kernel(const float* __restrict__ h,
                                                   const int* __restrict__ batch,
                                                   float* __restrict__ pooled,
                                                   float* __restrict__ counts) {
    const int node = (blockIdx.x * 256 + threadIdx.x) >> 5;   // one wave per node
    const int lane = threadIdx.x & 31;
    if (node >= N_NODES) return;
    const int gi = batch[node];
    const float* __restrict__ hp = h + (size_t)node * HIDDEN;
    float* __restrict__ pp = pooled + (size_t)gi * HIDDEN;
    for (int c = lane * 4; c < HIDDEN; c += 128) {
        const float4 v = *(const float4*)(hp + c);
        atomicAdd(pp + c + 0, v.x);
        atomicAdd(pp + c + 1, v.y);
        atomicAdd(pp + c + 2, v.z);
        atomicAdd(pp + c + 3, v.w);
    }
    if (lane == 0) atomicAdd(counts + gi, 1.0f);
}

// ---------------- head: logits = ReLU([pooled/cnt, g] @ hW1 + hb1) @ hW2 + hb2 ----------------
__global__ __launch_bounds__(HIDDEN) void head_kernel(const float* __restrict__ pooled,
                                                      const float* __restrict__ counts,
                                                      const float* __restrict__ g,
                                                      const float* __restrict__ hW1,
                                                      const float* __restrict__ hb1,
                                                      const float* __restrict__ hW2,
                                                      const float* __restrict__ hb2,
                                                      float* __restrict__ out) {
    const int gi = blockIdx.x;
    const int t  = threadIdx.x;              // hidden unit
    const float inv = 1.0f / fmaxf(counts[gi], 1.0f);
    float acc = hb1[t];
    for (int k = 0; k < HIDDEN; ++k)
        acc += (pooled[(size_t)gi * HIDDEN + k] * inv) * hW1[(size_t)k * HIDDEN + t];
    for (int k = 0; k < GLOB_DIM; ++k)
        acc += g[(size_t)gi * GLOB_DIM + k] * hW1[(size_t)(HIDDEN + k) * HIDDEN + t];
    acc = fmaxf(acc, 0.0f);

    __shared__ float red[HIDDEN];
    red[t] = acc * hW2[t];
    __syncthreads();
    for (int s = HIDDEN / 2; s > 0; s >>= 1) {
        if (t < s) red[t] += red[t + s];
        __syncthreads();
    }
    if (t == 0) out[gi] = red[0] + hb2[0];
}

extern "C" void kernel_launch(void* const* d_in, const int* in_sizes, int n_in,
                              void* d_out, int out_size, void* d_ws, size_t ws_size,
                              hipStream_t stream) {
    const float* x     = (const float*)d_in[0];
    const int*   ei    = (const int*)d_in[1];
    const int*   src   = ei;
    const int*   dst   = ei + N_EDGES;
    const int*   batch = (const int*)d_in[2];
    const float* g     = (const float*)d_in[3];
    const float *W1[4], *b1[4], *W2[4], *b2[4];
    for (int i = 0; i < 4; ++i) {
        W1[i] = (const float*)d_in[4 + 4 * i];
        b1[i] = (const float*)d_in[5 + 4 * i];
        W2[i] = (const float*)d_in[6 + 4 * i];
        b2[i] = (const float*)d_in[7 + 4 * i];
    }
    const float* eps = (const float*)d_in[20];
    const float* hW1 = (const float*)d_in[21];
    const float* hb1 = (const float*)d_in[22];
    const float* hW2 = (const float*)d_in[23];
    const float* hb2 = (const float*)d_in[24];
    float* out = (float*)d_out;

    const size_t S = (size_t)N_NODES * HIDDEN;     // elements per activation buffer
    float* bufA   = (float*)d_ws;
    float* bufB   = bufA + S;
    float* bufC   = bufB + S;
    float* pooled = bufC + S;
    float* counts = pooled + (size_t)N_GRAPHS * HIDDEN;

    const int ggrid = N_NODES / 16;                // 1250 blocks, 16x256 strip each
    const int scatter_blocks = (N_EDGES * 32 + 255) / 256;

    const float* h = x;
    for (int i = 0; i < 4; ++i) {
        const int Kin = (i == 0) ? NODE_DIM : HIDDEN;
        float* z = (i == 0) ? bufA : ((h == bufA) ? bufB : bufA);
        float* t = bufC;
        float* hnew = z;                           // z is dead once GEMM1 is done

        const int nz = N_NODES * Kin;
        zinit_kernel<<<(nz + 255) / 256, 256, 0, stream>>>(h, eps, i, z, nz);
        scatter_kernel<<<scatter_blocks, 256, 0, stream>>>(h, src, dst, z, N_EDGES, Kin);

        if (i == 0) {
            gemm_relu_wmma<NODE_DIM, false><<<ggrid, 128, 0, stream>>>(z, W1[i], b1[i], nullptr, t);
            gemm_relu_wmma<HIDDEN, false><<<ggrid, 128, 0, stream>>>(t, W2[i], b2[i], nullptr, hnew);
        } else {
            gemm_relu_wmma<HIDDEN, false><<<ggrid, 128, 0, stream>>>(z, W1[i], b1[i], nullptr, t);
            gemm_relu_wmma<HIDDEN, true><<<ggrid, 128, 0, stream>>>(t, W2[i], b2[i], h, hnew);
        }
        h = hnew;
    }

    const int npool = N_GRAPHS * HIDDEN + N_GRAPHS;    // pooled + counts, contiguous
    zero_kernel<<<(npool + 255) / 256, 256, 0, stream>>>(pooled, npool);
    pool_kernel<<<(N_NODES * 32 + 255) / 256, 256, 0, stream>>>(h, batch, pooled, counts);
    head_kernel<<<N_GRAPHS, HIDDEN, 0, stream>>>(pooled, counts, g, hW1, hb1, hW2, hb2, out);
}